// MSTDependencyParser_68762426409879
// MI455X (gfx1250) — compile-verified
//
#include <hip/hip_runtime.h>
#include <hip/hip_bf16.h>
#include <math.h>

typedef __attribute__((ext_vector_type(2))) float v2f;
typedef __attribute__((ext_vector_type(8))) float v8f;

#define N_TOK 1024
#define WD    300
#define PD    25
#define IN0   325     // WD + PD
#define H     125
#define G4    500     // 4*H
#define IN1   250     // 2*H
#define FC    100

// ---------------- embedding concat: seq[i] = [words[i], pos_table[idx[i]]] ----------------
__global__ void k_concat(const float* __restrict__ words,
                         const float* __restrict__ pos_table,
                         const int*   __restrict__ idx,
                         float* __restrict__ seq)
{
    int i = blockIdx.x;
    int p = idx[i];
    for (int k = threadIdx.x; k < IN0; k += blockDim.x) {
        float v = (k < WD) ? words[i * WD + k] : pos_table[p * PD + (k - WD)];
        seq[i * IN0 + k] = v;
    }
}

// ------- WMMA f32 GEMM: C[M,N] = A[M,K] * W[N,K]^T + b1[N] + b2[N]; blockIdx.z = dir -------
// one wave32 per 16x16 output tile; V_WMMA_F32_16X16X4_F32 accumulation over K
__global__ void k_gemm_wmma(const float* __restrict__ A,
                            const float* __restrict__ Wbase,
                            const float* __restrict__ b1base,
                            const float* __restrict__ b2base,
                            float* __restrict__ Cbase,
                            int M, int N, int K, int lda, int ldw, int ldc,
                            long wDirStride, long bDirStride, long cDirStride)
{
    const int dir = blockIdx.z;
    const float* Wm = Wbase + (long)dir * wDirStride;
    const float* b1 = b1base ? (b1base + (long)dir * bDirStride) : nullptr;
    const float* b2 = b2base ? (b2base + (long)dir * bDirStride) : nullptr;
    float* C = Cbase + (long)dir * cDirStride;

    const int lane = threadIdx.x;            // 0..31
    const int mr   = lane & 15;              // M (for A) / N (for B) index
    const int kb   = (lane >> 4) << 1;       // K sub-offset: lanes 0-15 -> {0,1}, 16-31 -> {2,3}
    const int tm   = blockIdx.y * 16;
    const int tn   = blockIdx.x * 16;

    const int m    = tm + mr;                            // M always multiple of 16 here
    const int ncol = tn + mr;
    const int ncl  = (ncol < N) ? ncol : 0;              // clamp OOB W rows (cols never stored)

    const float* arow = A  + (long)m   * lda;
    const float* wrow = Wm + (long)ncl * ldw;

    v8f acc = {0.f, 0.f, 0.f, 0.f, 0.f, 0.f, 0.f, 0.f};

    const int K4 = K & ~3;
    for (int k = 0; k < K4; k += 4) {
        v2f av, bv;
        av.x = arow[k + kb];     av.y = arow[k + kb + 1];
        bv.x = wrow[k + kb];     bv.y = wrow[k + kb + 1];
        acc = __builtin_amdgcn_wmma_f32_16x16x4_f32(false, av, false, bv,
                                                    (short)0, acc, false, false);
    }
    if (K4 < K) {  // guarded K tail
        v2f av, bv;
        av.x = (K4 + kb     < K) ? arow[K4 + kb]     : 0.f;
        av.y = (K4 + kb + 1 < K) ? arow[K4 + kb + 1] : 0.f;
        bv.x = (K4 + kb     < K) ? wrow[K4 + kb]     : 0.f;
        bv.y = (K4 + kb + 1 < K) ? wrow[K4 + kb + 1] : 0.f;
        acc = __builtin_amdgcn_wmma_f32_16x16x4_f32(false, av, false, bv,
                                                    (short)0, acc, false, false);
    }

    // D layout: VGPR r -> row tm + r + (lane>=16 ? 8 : 0), col tn + (lane&15)
    if (ncol < N) {
        float bias = (b1 ? b1[ncol] : 0.f) + (b2 ? b2[ncol] : 0.f);
        const int rbase = tm + ((lane >> 4) << 3);
        #pragma unroll
        for (int r = 0; r < 8; ++r)
            C[(long)(rbase + r) * ldc + ncol] = acc[r] + bias;
    }
}

// -------- sequential LSTM scan, one workgroup per direction; Whh cached in LDS --------
// xg: [2][n][500] precomputed input gates (incl. biases); v: [n][250] output (dir-offset cols)
__global__ void k_lstm_scan(const float* __restrict__ xg,
                            const float* __restrict__ Whh,   // [2][500][125]
                            float* __restrict__ v, int n)
{
    extern __shared__ float smem[];
    float* Wl = smem;                 // 62500 floats = 250 KB (fits 320 KB LDS)
    float* h  = smem + G4 * H;        // 128
    float* c  = h + 128;              // 128
    float* g  = c + 128;              // 512
    const int dir = blockIdx.x;
    const int tid = threadIdx.x;

    const float* Wg  = Whh + (long)dir * G4 * H;
    const float* xgd = xg  + (long)dir * (long)n * G4;

    for (int i = tid; i < G4 * H; i += blockDim.x) Wl[i] = Wg[i];
    if (tid < H) { h[tid] = 0.f; c[tid] = 0.f; }
    __syncthreads();

    for (int s = 0; s < n; ++s) {
        const int t = dir ? (n - 1 - s) : s;
        if (tid < G4) {
            const float* wr = Wl + tid * H;      // stride 125 -> bank-conflict-free
            float acc = xgd[(long)t * G4 + tid];
            #pragma unroll 5
            for (int k = 0; k < H; ++k) acc += h[k] * wr[k];
            g[tid] = acc;
        }
        __syncthreads();
        if (tid < H) {  // PyTorch gate order i, f, g, o
            float gi = g[tid], gf = g[H + tid], gc = g[2 * H + tid], go = g[3 * H + tid];
            float si = 1.f / (1.f + __expf(-gi));
            float sf = 1.f / (1.f + __expf(-gf));
            float so = 1.f / (1.f + __expf(-go));
            float cc = sf * c[tid] + si * tanhf(gc);
            float hh = so * tanhf(cc);
            c[tid] = cc; h[tid] = hh;
            v[(long)t * IN1 + dir * H + tid] = hh;
        }
        __syncthreads();
    }
}

// -------- pairwise scorer: out[i,j] = fc2_b + sum_k fc2_w[k]*tanh(b[i,k]+a[j,k]) --------
// (a already contains fc1_b). 16x16 output tile per block; a/b rows staged in LDS.
__global__ void k_pairwise(const float* __restrict__ a,
                           const float* __restrict__ b,
                           const float* __restrict__ w2,
                           const float* __restrict__ bb2,
                           float* __restrict__ out, int n)
{
    __shared__ float at[16][FC];
    __shared__ float bt[16][FC];
    __shared__ float w[FC];
    const int tj = threadIdx.x, ti = threadIdx.y;
    const int tid = ti * 16 + tj;
    const int i0 = blockIdx.y * 16, j0 = blockIdx.x * 16;

    for (int idx = tid; idx < 16 * FC; idx += 256) {
        int r = idx / FC, k = idx % FC;
        at[r][k] = a[(long)(j0 + r) * FC + k];
        bt[r][k] = b[(long)(i0 + r) * FC + k];
    }
    if (tid < FC) w[tid] = w2[tid];
    __syncthreads();

    float s = bb2[0];
    #pragma unroll 4
    for (int k = 0; k < FC; ++k)
        s += w[k] * tanhf(bt[ti][k] + at[tj][k]);
    out[(long)(i0 + ti) * n + (j0 + tj)] = s;
}

extern "C" void kernel_launch(void* const* d_in, const int* in_sizes, int n_in,
                              void* d_out, int out_size, void* d_ws, size_t ws_size,
                              hipStream_t stream)
{
    const float* words     = (const float*)d_in[0];
    const float* pos_table = (const float*)d_in[1];
    const float* Wih0      = (const float*)d_in[2];
    const float* Whh0      = (const float*)d_in[3];
    const float* bih0      = (const float*)d_in[4];
    const float* bhh0      = (const float*)d_in[5];
    const float* Wih1      = (const float*)d_in[6];
    const float* Whh1      = (const float*)d_in[7];
    const float* bih1      = (const float*)d_in[8];
    const float* bhh1      = (const float*)d_in[9];
    const float* fc1_w     = (const float*)d_in[10];  // [100][500]
    const float* fc1_b     = (const float*)d_in[11];
    const float* fc2_w     = (const float*)d_in[12];  // [1][100]
    const float* fc2_b     = (const float*)d_in[13];
    const int*   idx       = (const int*)d_in[14];
    float* out = (float*)d_out;

    float* ws  = (float*)d_ws;                 // ~11.8 MB total
    float* seq = ws;                           // 1024*325
    float* xg0 = seq + (long)N_TOK * IN0;      // 2*1024*500
    float* v0  = xg0 + 2L * N_TOK * G4;        // 1024*250
    float* xg1 = v0  + (long)N_TOK * IN1;      // 2*1024*500
    float* v1  = xg1 + 2L * N_TOK * G4;        // 1024*250
    float* av  = v1  + (long)N_TOK * IN1;      // 1024*100
    float* bv  = av  + (long)N_TOK * FC;       // 1024*100

    // 1) embedding concat
    k_concat<<<N_TOK, 128, 0, stream>>>(words, pos_table, idx, seq);

    // 2) xg0[d] = seq @ Wih0[d]^T + bih0[d] + bhh0[d]
    dim3 g0((G4 + 15) / 16, N_TOK / 16, 2);
    k_gemm_wmma<<<g0, 32, 0, stream>>>(seq, Wih0, bih0, bhh0, xg0,
                                       N_TOK, G4, IN0, IN0, IN0, G4,
                                       (long)G4 * IN0, G4, (long)N_TOK * G4);

    // 3) layer-0 recurrence (fwd/bwd on separate WGPs, Whh in LDS)
    size_t lds_bytes = (size_t)(G4 * H + 128 + 128 + 512) * sizeof(float);
    k_lstm_scan<<<2, 512, lds_bytes, stream>>>(xg0, Whh0, v0, N_TOK);

    // 4) xg1[d] = v0 @ Wih1[d]^T + bih1[d] + bhh1[d]
    k_gemm_wmma<<<g0, 32, 0, stream>>>(v0, Wih1, bih1, bhh1, xg1,
                                       N_TOK, G4, IN1, IN1, IN1, G4,
                                       (long)G4 * IN1, G4, (long)N_TOK * G4);

    // 5) layer-1 recurrence
    k_lstm_scan<<<2, 512, lds_bytes, stream>>>(xg1, Whh1, v1, N_TOK);

    // 6) a = v1 @ fc1_w[:, :250]^T + fc1_b ; b = v1 @ fc1_w[:, 250:]^T
    dim3 gfc((FC + 15) / 16, N_TOK / 16, 1);
    k_gemm_wmma<<<gfc, 32, 0, stream>>>(v1, fc1_w, fc1_b, nullptr, av,
                                        N_TOK, FC, IN1, IN1, 2 * IN1, FC, 0, 0, 0);
    k_gemm_wmma<<<gfc, 32, 0, stream>>>(v1, fc1_w + IN1, nullptr, nullptr, bv,
                                        N_TOK, FC, IN1, IN1, 2 * IN1, FC, 0, 0, 0);

    // 7) pairwise tanh reduction -> scores [1024,1024]
    dim3 gp(N_TOK / 16, N_TOK / 16, 1);
    k_pairwise<<<gp, dim3(16, 16, 1), 0, stream>>>(av, bv, fc2_w, fc2_b, out, N_TOK);
}